// GraphSAGE_317827580339
// MI455X (gfx1250) — compile-verified
//
#include <hip/hip_runtime.h>
#include <hip/hip_bf16.h>

// ---------------------------------------------------------------------------
// GraphSAGE on MI455X (gfx1250): fp32 WMMA GEMMs + atomic-scatter aggregation.
// ---------------------------------------------------------------------------

typedef float v2f __attribute__((ext_vector_type(2)));
typedef float v8f __attribute__((ext_vector_type(8)));

#define DIN  128
#define DH   256
#define DOUT 128

// ----------------------------- utility kernels -----------------------------

__global__ void zero_kernel(float* __restrict__ p, long n) {
    long i = (long)blockIdx.x * blockDim.x + threadIdx.x;
    long stride = (long)gridDim.x * blockDim.x;
    for (; i < n; i += stride) p[i] = 0.0f;
}

__global__ void count_kernel(const int* __restrict__ dst, float* __restrict__ deg, int E) {
    int e = blockIdx.x * blockDim.x + threadIdx.x;
    if (e < E) atomicAdd(&deg[dst[e]], 1.0f);
}

__global__ void recip_kernel(float* __restrict__ deg, int n) {
    int i = blockIdx.x * blockDim.x + threadIdx.x;
    if (i < n) deg[i] = 1.0f / fmaxf(deg[i], 1.0f);
}

// Gather x[src] (float4) and scatter-add into agg[dst] (4x global_atomic_add_f32).
__global__ void scatter_kernel(const float* __restrict__ x,
                               const int* __restrict__ src,
                               const int* __restrict__ dst,
                               float* __restrict__ agg,
                               int E, int d) {
    long t = (long)blockIdx.x * blockDim.x + threadIdx.x;
    int chunks = d >> 2;                     // float4 chunks per row
    long total = (long)E * chunks;
    if (t >= total) return;
    int e = (int)(t / chunks);
    int f = (int)(t % chunks) * 4;
    int s = src[e];
    int dd = dst[e];
    const float4 v = *(const float4*)(x + (size_t)s * d + f);
    float* a = agg + (size_t)dd * d + f;
    atomicAdd(a + 0, v.x);
    atomicAdd(a + 1, v.y);
    atomicAdd(a + 2, v.z);
    atomicAdd(a + 3, v.w);
}

// ----------------------------- fused WMMA GEMM -----------------------------
// One wave (32 threads) computes a 16x32 output tile (two 16x16 WMMA tiles,
// sharing the A fragments) of:
//   Out1 = (invDeg ∘ Agg) @ Wl + bl + X @ Wr
//   Out2 = X @ Wsk + bsk                       (HAS_SKIP instantiation only)
//
// V_WMMA_F32_16X16X4_F32 fragment layout (ISA 7.12.2, wave32):
//   A (16x4):  lane L<16 holds row M=L, K={kk,kk+1}; lane L>=16 holds row
//              M=L-16, K={kk+2,kk+3}.
//   B (4x16):  VGPR j: K=kk+2*half+j, N=lane%16.
//   C/D:       VGPR i: row M = i + 8*half, col N = lane%16.
//
// B addressing: uniform row pointer (scalar, stepped by 4*C per iteration) +
// loop-invariant 32-bit lane offset -> saddr+voffset global loads, no per-lane
// 64-bit pointer updates in the loop. HAS_SKIP is a template parameter so no
// EXEC-mask games happen around WMMA (ISA: EXEC must be all ones for WMMA).
template <bool HAS_SKIP>
__global__ __launch_bounds__(32)
void sage_gemm_kernel(const float* __restrict__ Agg,
                      const float* __restrict__ X,
                      const float* __restrict__ invDeg,
                      const float* __restrict__ Wl, const float* __restrict__ bl,
                      const float* __restrict__ Wr,
                      const float* __restrict__ Wsk, const float* __restrict__ bsk,
                      float* __restrict__ Out1, float* __restrict__ Out2,
                      int Nn, int K, int C) {
    const int lane = threadIdx.x & 31;
    const int half = lane >> 4;         // 0 or 1
    const int l16  = lane & 15;
    const int rowBase = blockIdx.x * 16;
    const int colBase = blockIdx.y * 32;   // two 16-wide column tiles per wave

    int aRow = rowBase + l16;
    if (aRow >= Nn) aRow = Nn - 1;      // clamp (keeps EXEC all-ones)
    const int col0 = colBase + l16;
    const int col1 = col0 + 16;

    const float inv = invDeg[aRow];

    // Divergent, loop-invariant 32-bit offsets (element units).
    const int bOff0 = 2 * half * C + col0;   // B fragment lane offset, tile 0
    const int bOff1 = bOff0 + 16;            // tile 1
    const float* aggRow = Agg + (size_t)aRow * K + 2 * half;
    const float* xRow   = X   + (size_t)aRow * K + 2 * half;

    // Uniform (scalar) row pointers, stepped by 4 rows per K-step.
    const float* wl = Wl;
    const float* wr = Wr;
    const float* ws = HAS_SKIP ? Wsk : nullptr;

    v8f accL0 = {}, accL1 = {};
    v8f accR0 = {}, accR1 = {};
    v8f accS0 = {}, accS1 = {};

    for (int kk = 0; kk < K; kk += 4) {
        v2f aA = { aggRow[kk] * inv, aggRow[kk + 1] * inv };
        v2f aX = { xRow[kk],         xRow[kk + 1]         };

        v2f bL0 = { wl[bOff0], wl[bOff0 + C] };
        v2f bL1 = { wl[bOff1], wl[bOff1 + C] };
        v2f bR0 = { wr[bOff0], wr[bOff0 + C] };
        v2f bR1 = { wr[bOff1], wr[bOff1 + C] };

        accL0 = __builtin_amdgcn_wmma_f32_16x16x4_f32(false, aA, false, bL0,
                                                      (short)0, accL0, false, false);
        accL1 = __builtin_amdgcn_wmma_f32_16x16x4_f32(false, aA, false, bL1,
                                                      (short)0, accL1, false, false);
        accR0 = __builtin_amdgcn_wmma_f32_16x16x4_f32(false, aX, false, bR0,
                                                      (short)0, accR0, false, false);
        accR1 = __builtin_amdgcn_wmma_f32_16x16x4_f32(false, aX, false, bR1,
                                                      (short)0, accR1, false, false);
        if constexpr (HAS_SKIP) {
            v2f bS0 = { ws[bOff0], ws[bOff0 + C] };
            v2f bS1 = { ws[bOff1], ws[bOff1 + C] };
            accS0 = __builtin_amdgcn_wmma_f32_16x16x4_f32(false, aX, false, bS0,
                                                          (short)0, accS0, false, false);
            accS1 = __builtin_amdgcn_wmma_f32_16x16x4_f32(false, aX, false, bS1,
                                                          (short)0, accS1, false, false);
            ws += 4 * (size_t)C;
        }
        wl += 4 * (size_t)C;
        wr += 4 * (size_t)C;
    }

    const float biasL0 = bl[col0];
    const float biasL1 = bl[col1];
#pragma unroll
    for (int i = 0; i < 8; ++i) {
        int r = rowBase + i + 8 * half;
        if (r < Nn) {
            Out1[(size_t)r * C + col0] = accL0[i] + accR0[i] + biasL0;
            Out1[(size_t)r * C + col1] = accL1[i] + accR1[i] + biasL1;
        }
    }
    if constexpr (HAS_SKIP) {
        const float biasS0 = bsk[col0];
        const float biasS1 = bsk[col1];
#pragma unroll
        for (int i = 0; i < 8; ++i) {
            int r = rowBase + i + 8 * half;
            if (r < Nn) {
                Out2[(size_t)r * C + col0] = accS0[i] + biasS0;
                Out2[(size_t)r * C + col1] = accS1[i] + biasS1;
            }
        }
    }
}

// --------------------- LayerNorm + ReLU + skip epilogue ---------------------
// One wave per row (C == 256 -> 8 values per lane), wave32 shfl-xor reduction.
__global__ __launch_bounds__(256)
void ln_relu_skip_kernel(const float* __restrict__ conv,
                         const float* __restrict__ skip,
                         const float* __restrict__ g,
                         const float* __restrict__ be,
                         float* __restrict__ out,
                         int Nn) {
    const int wave = threadIdx.x >> 5;
    const int lane = threadIdx.x & 31;
    const int row = blockIdx.x * 8 + wave;
    if (row >= Nn) return;

    const float* cr = conv + (size_t)row * DH;
    float v[8];
    float s = 0.0f, ss = 0.0f;
#pragma unroll
    for (int j = 0; j < 8; ++j) {
        float x = cr[lane + 32 * j];
        v[j] = x;
        s += x;
        ss += x * x;
    }
#pragma unroll
    for (int off = 16; off > 0; off >>= 1) {
        s  += __shfl_xor(s, off, 32);
        ss += __shfl_xor(ss, off, 32);
    }
    const float mu  = s * (1.0f / DH);
    const float var = fmaxf(ss * (1.0f / DH) - mu * mu, 0.0f);
    const float rstd = rsqrtf(var + 1e-5f);

    const float* sr = skip + (size_t)row * DH;
    float* orow = out + (size_t)row * DH;
#pragma unroll
    for (int j = 0; j < 8; ++j) {
        int c = lane + 32 * j;
        float y = (v[j] - mu) * rstd * g[c] + be[c];
        y = fmaxf(y, 0.0f);
        orow[c] = y + sr[c];
    }
}

// --------------------------------- launch ----------------------------------

static inline unsigned ceil_div_u(long a, long b) { return (unsigned)((a + b - 1) / b); }

extern "C" void kernel_launch(void* const* d_in, const int* in_sizes, int n_in,
                              void* d_out, int out_size, void* d_ws, size_t ws_size,
                              hipStream_t stream) {
    // Input order per setup_inputs():
    const float* x    = (const float*)d_in[0];
    const int*   ei   = (const int*)d_in[1];
    const float* W1l  = (const float*)d_in[2];
    const float* b1l  = (const float*)d_in[3];
    const float* W1r  = (const float*)d_in[4];
    const float* g1   = (const float*)d_in[5];
    const float* be1  = (const float*)d_in[6];
    const float* Wsk  = (const float*)d_in[7];
    const float* bsk  = (const float*)d_in[8];
    const float* W2l  = (const float*)d_in[9];
    const float* b2l  = (const float*)d_in[10];
    const float* W2r  = (const float*)d_in[11];
    const float* g2   = (const float*)d_in[12];
    const float* be2  = (const float*)d_in[13];
    const float* W3l  = (const float*)d_in[14];
    const float* b3l  = (const float*)d_in[15];
    const float* W3r  = (const float*)d_in[16];

    const int Nn = in_sizes[0] / DIN;
    const int E  = in_sizes[1] / 2;
    const int* src = ei;        // edge_index[0]
    const int* dst = ei + E;    // edge_index[1]

    // Workspace layout (floats).
    float* ws   = (float*)d_ws;
    size_t o = 0;
    float* deg  = ws + o; o += (size_t)Nn;            // degree -> inv degree
    float* agg  = ws + o; o += (size_t)Nn * DH;
    float* conv = ws + o; o += (size_t)Nn * DH;
    float* skip = ws + o; o += (size_t)Nn * DH;
    float* h    = ws + o; o += (size_t)Nn * DH;
    float* h2   = ws + o; o += (size_t)Nn * DH;
    (void)ws_size; (void)n_in; (void)out_size;

    const int ZB = 256;
    const unsigned zgrid = 2048;   // grid-stride zeroing

    // ---- degree / inverse degree (shared by all 3 layers) ----
    zero_kernel<<<ceil_div_u(Nn, ZB), ZB, 0, stream>>>(deg, (long)Nn);
    count_kernel<<<ceil_div_u(E, ZB), ZB, 0, stream>>>(dst, deg, E);
    recip_kernel<<<ceil_div_u(Nn, ZB), ZB, 0, stream>>>(deg, Nn);

    // ---------------- layer 0: DIN -> DH, LN+ReLU, projected skip ----------
    zero_kernel<<<zgrid, ZB, 0, stream>>>(agg, (long)Nn * DIN);
    scatter_kernel<<<ceil_div_u((long)E * (DIN / 4), ZB), ZB, 0, stream>>>(
        x, src, dst, agg, E, DIN);
    {
        dim3 grid(ceil_div_u(Nn, 16), DH / 32);
        sage_gemm_kernel<true><<<grid, 32, 0, stream>>>(
            agg, x, deg, W1l, b1l, W1r, Wsk, bsk, conv, skip, Nn, DIN, DH);
    }
    ln_relu_skip_kernel<<<ceil_div_u(Nn, 8), 256, 0, stream>>>(
        conv, skip, g1, be1, h, Nn);

    // ---------------- layer 1: DH -> DH, LN+ReLU, identity residual --------
    zero_kernel<<<zgrid, ZB, 0, stream>>>(agg, (long)Nn * DH);
    scatter_kernel<<<ceil_div_u((long)E * (DH / 4), ZB), ZB, 0, stream>>>(
        h, src, dst, agg, E, DH);
    {
        dim3 grid(ceil_div_u(Nn, 16), DH / 32);
        sage_gemm_kernel<false><<<grid, 32, 0, stream>>>(
            agg, h, deg, W2l, b2l, W2r, nullptr, nullptr, conv, nullptr, Nn, DH, DH);
    }
    ln_relu_skip_kernel<<<ceil_div_u(Nn, 8), 256, 0, stream>>>(
        conv, h, g2, be2, h2, Nn);

    // ---------------- layer 2: DH -> DOUT, plain conv into d_out -----------
    zero_kernel<<<zgrid, ZB, 0, stream>>>(agg, (long)Nn * DH);
    scatter_kernel<<<ceil_div_u((long)E * (DH / 4), ZB), ZB, 0, stream>>>(
        h2, src, dst, agg, E, DH);
    {
        dim3 grid(ceil_div_u(Nn, 16), DOUT / 32);
        sage_gemm_kernel<false><<<grid, 32, 0, stream>>>(
            agg, h2, deg, W3l, b3l, W3r, nullptr, nullptr, (float*)d_out, nullptr,
            Nn, DH, DOUT);
    }
}